// StructureAttention_45329084842167
// MI455X (gfx1250) — compile-verified
//
#include <hip/hip_runtime.h>
#include <hip/hip_bf16.h>

// Problem constants (match reference)
#define B_   16
#define N_   2048
#define DG_  512
#define DH_  512
#define IN_  1024
#define OUT_ 512
#define KTOP 512   // floor(2048 * 0.25)

typedef float v2f __attribute__((ext_vector_type(2)));
typedef float v4f __attribute__((ext_vector_type(4)));
typedef float v8f __attribute__((ext_vector_type(8)));

// CDNA5 native f32 matrix op: D(16x16) = A(16x4) * B(4x16) + C
static __device__ __forceinline__ v8f wmma_f32(v2f a, v2f b, v8f c) {
  return __builtin_amdgcn_wmma_f32_16x16x4_f32(false, a, false, b, (short)0, c,
                                               false, false);
}

// order-preserving float<->uint keys (for atomicMin over signed floats)
static __device__ __forceinline__ unsigned fkey(float f) {
  unsigned u = __float_as_uint(f);
  return u ^ ((unsigned)((int)u >> 31) | 0x80000000u);
}
static __device__ __forceinline__ float funkey(unsigned k) {
  unsigned m = (k & 0x80000000u) ? 0x80000000u : 0xFFFFFFFFu;
  return __uint_as_float(k ^ m);
}

__global__ void sa_init_kernel(unsigned* __restrict__ minkeys) {
  if (threadIdx.x < B_) minkeys[threadIdx.x] = 0xFFFFFFFFu;  // +max key
}

// ---------------------------------------------------------------------------
// Kernel 1: e1 = (concat(ge,h)*mask) @ W1 + b1 ; e2 likewise with W2,b2
// One wave -> one 16x16 tile of BOTH e1 and e2 (A operand reused).
// waves = B * (N/16) * (OUT/16) = 65536 ; 4 waves/block -> 16384 blocks
// e1/e2 are reused heavily by kernel 2 -> regular-temporal stores (L2 keeps
// them; 128 MB fits in the 192 MB global L2).
// ---------------------------------------------------------------------------
__global__ __launch_bounds__(128) void sa_embed_kernel(
    const float* __restrict__ ge, const float* __restrict__ mask,
    const float* __restrict__ hs, const float* __restrict__ W1,
    const float* __restrict__ b1, const float* __restrict__ W2,
    const float* __restrict__ b2, float* __restrict__ e1,
    float* __restrict__ e2) {
  const int lane = threadIdx.x & 31;
  const int gw   = blockIdx.x * (blockDim.x >> 5) + (threadIdx.x >> 5);
  const int b    = gw >> 12;          // 4096 tiles per batch
  const int t    = gw & 4095;
  const int tm   = (t >> 5) << 4;     // row tile (128 tiles)
  const int tn   = (t & 31) << 4;     // col tile (32 tiles)
  const int l15  = lane & 15;
  const int kh   = (lane >> 4) << 1;  // K sub-offset: 0 (lanes 0-15) / 2 (16-31)
  const int row  = tm + l15;          // A row carried by this lane
  const int col  = tn + l15;          // B/C column carried by this lane

  const float mrow = mask[b * N_ + row];
  const float* __restrict__ gerow = ge + ((size_t)(b * N_ + row)) * DG_;
  const float* __restrict__ hrow  = hs + (size_t)b * DH_;

  v8f acc1 = {0.f, 0.f, 0.f, 0.f, 0.f, 0.f, 0.f, 0.f};
  v8f acc2 = {0.f, 0.f, 0.f, 0.f, 0.f, 0.f, 0.f, 0.f};

  // K in [0, 512): A from graph_embed
#pragma unroll 4
  for (int k0 = 0; k0 < DG_; k0 += 4) {
    const int kk = k0 + kh;
    v2f a = *(const v2f*)(gerow + kk);
    a *= mrow;
    v2f w1v, w2v;
    w1v.x = W1[(size_t)kk * OUT_ + col];
    w1v.y = W1[(size_t)(kk + 1) * OUT_ + col];
    w2v.x = W2[(size_t)kk * OUT_ + col];
    w2v.y = W2[(size_t)(kk + 1) * OUT_ + col];
    acc1 = wmma_f32(a, w1v, acc1);
    acc2 = wmma_f32(a, w2v, acc2);
  }
  // K in [512, 1024): A from hidden_state (broadcast over rows, still masked)
#pragma unroll 4
  for (int k0 = DG_; k0 < IN_; k0 += 4) {
    const int kk = k0 + kh;
    v2f a = *(const v2f*)(hrow + (kk - DG_));
    a *= mrow;
    v2f w1v, w2v;
    w1v.x = W1[(size_t)kk * OUT_ + col];
    w1v.y = W1[(size_t)(kk + 1) * OUT_ + col];
    w2v.x = W2[(size_t)kk * OUT_ + col];
    w2v.y = W2[(size_t)(kk + 1) * OUT_ + col];
    acc1 = wmma_f32(a, w1v, acc1);
    acc2 = wmma_f32(a, w2v, acc2);
  }

  // Bias (same N column across all 8 C/D elements of a lane) + store.
  const float bias1 = b1[col];
  const float bias2 = b2[col];
  const int rbase = tm + ((lane >> 4) << 3);  // +8 rows for upper half-wave
#pragma unroll
  for (int j = 0; j < 8; ++j) {
    const size_t idx = ((size_t)(b * N_ + rbase + j)) * OUT_ + col;
    e1[idx] = acc1[j] + bias1;
    e2[idx] = acc2[j] + bias2;
  }
}

// ---------------------------------------------------------------------------
// Kernel 2: Adj[b] = e1[b] @ e2[b]^T  (K=512), plus per-batch global min.
// One wave -> 32x64 output tile: 6 float2 loads : 8 WMMAs per K-step
// (8 v8f accumulators = 64 VGPRs). Adj is streamed out with NT stores so the
// 268 MB result does not evict the L2-resident e1/e2 operands.
// waves = 16 * (2048/32) * (2048/64) = 32768 ; 4 waves/block -> 8192 blocks
// ---------------------------------------------------------------------------
__global__ __launch_bounds__(128) void sa_adj_kernel(
    const float* __restrict__ e1, const float* __restrict__ e2,
    float* __restrict__ Adj, unsigned* __restrict__ minkeys) {
  const int lane = threadIdx.x & 31;
  const int gw   = blockIdx.x * (blockDim.x >> 5) + (threadIdx.x >> 5);
  const int b    = gw >> 11;          // 2048 tiles per batch (64 x 32)
  const int t    = gw & 2047;
  const int tm   = (t >> 5) << 5;     // 64 row-tiles of 32
  const int tn   = (t & 31) << 6;     // 32 col-tiles of 64
  const int l15  = lane & 15;
  const int kh   = (lane >> 4) << 1;

  const float* __restrict__ a0p = e1 + ((size_t)(b * N_ + tm + l15)) * OUT_ + kh;
  const float* __restrict__ a1p = a0p + (size_t)16 * OUT_;
  const float* __restrict__ b0p = e2 + ((size_t)(b * N_ + tn + l15)) * OUT_ + kh;
  const float* __restrict__ b1p = b0p + (size_t)16 * OUT_;
  const float* __restrict__ b2p = b0p + (size_t)32 * OUT_;
  const float* __restrict__ b3p = b0p + (size_t)48 * OUT_;

  v8f acc00 = {0.f, 0.f, 0.f, 0.f, 0.f, 0.f, 0.f, 0.f};
  v8f acc01 = acc00, acc02 = acc00, acc03 = acc00;
  v8f acc10 = acc00, acc11 = acc00, acc12 = acc00, acc13 = acc00;

#pragma unroll 2
  for (int k0 = 0; k0 < OUT_; k0 += 4) {
    const v2f a0  = *(const v2f*)(a0p + k0);
    const v2f a1  = *(const v2f*)(a1p + k0);
    const v2f bb0 = *(const v2f*)(b0p + k0);
    const v2f bb1 = *(const v2f*)(b1p + k0);
    const v2f bb2 = *(const v2f*)(b2p + k0);
    const v2f bb3 = *(const v2f*)(b3p + k0);
    acc00 = wmma_f32(a0, bb0, acc00);
    acc01 = wmma_f32(a0, bb1, acc01);
    acc02 = wmma_f32(a0, bb2, acc02);
    acc03 = wmma_f32(a0, bb3, acc03);
    acc10 = wmma_f32(a1, bb0, acc10);
    acc11 = wmma_f32(a1, bb1, acc11);
    acc12 = wmma_f32(a1, bb2, acc12);
    acc13 = wmma_f32(a1, bb3, acc13);
  }

  const int rposBase = (lane >> 4) << 3;
  float mloc = __int_as_float(0x7F800000);  // +inf

#define SA_STORE_TILE(ACC, IM, JN)                                          \
  do {                                                                      \
    const int rowb = tm + (IM)*16 + rposBase;                               \
    const int coln = tn + (JN)*16 + l15;                                    \
    _Pragma("unroll") for (int j = 0; j < 8; ++j) {                         \
      const float v = (ACC)[j];                                             \
      __builtin_nontemporal_store(                                          \
          v, &Adj[((size_t)b * N_ + (size_t)(rowb + j)) * N_ + coln]);      \
      mloc = fminf(mloc, v);                                                \
    }                                                                       \
  } while (0)

  SA_STORE_TILE(acc00, 0, 0);
  SA_STORE_TILE(acc01, 0, 1);
  SA_STORE_TILE(acc02, 0, 2);
  SA_STORE_TILE(acc03, 0, 3);
  SA_STORE_TILE(acc10, 1, 0);
  SA_STORE_TILE(acc11, 1, 1);
  SA_STORE_TILE(acc12, 1, 2);
  SA_STORE_TILE(acc13, 1, 3);
#undef SA_STORE_TILE

  // wave32 min-reduce, then atomic per-batch min on order-preserving keys
#pragma unroll
  for (int off = 16; off >= 1; off >>= 1)
    mloc = fminf(mloc, __shfl_xor(mloc, off, 32));
  if (lane == 0) atomicMin(&minkeys[b], fkey(mloc));
}

// ---------------------------------------------------------------------------
// Kernel 3: per-row radix-select of the 512th-largest of (Adj - min)*mask_row,
// then row-sum over kept entries -> d_inv_sqrt. One 256-thread block per row.
// Adj is streamed (read once here) -> NT loads.
// ---------------------------------------------------------------------------
__global__ __launch_bounds__(256) void sa_topk_kernel(
    const float* __restrict__ Adj, const float* __restrict__ mask,
    const unsigned* __restrict__ minkeys, unsigned* __restrict__ thr,
    float* __restrict__ dinv) {
  __shared__ unsigned hist[256];
  __shared__ float fsum[256];
  __shared__ unsigned selB, selR;

  const int tid    = threadIdx.x;
  const int rowIdx = blockIdx.x;       // b*N + r
  const int b      = rowIdx >> 11;
  const float minv = funkey(minkeys[b]);
  const float mr   = mask[rowIdx];
  const float* __restrict__ row = Adj + (size_t)rowIdx * N_;

  // x >= 0 always (Adj >= min, mask >= 0) -> float bits are order-preserving
  float x[8];
  unsigned bits[8];
#pragma unroll
  for (int i = 0; i < 8; ++i) {
    const float v  = __builtin_nontemporal_load(row + i * 256 + tid);
    const float xv = (v - minv) * mr;
    x[i]    = xv;
    bits[i] = __float_as_uint(xv);
  }

  unsigned prefix = 0;
  unsigned remaining = KTOP;
  for (int shift = 24; shift >= 0; shift -= 8) {
    hist[tid] = 0;
    __syncthreads();
    const unsigned pmask = (shift == 24) ? 0u : (0xFFFFFFFFu << (shift + 8));
#pragma unroll
    for (int i = 0; i < 8; ++i)
      if ((bits[i] & pmask) == prefix)
        atomicAdd(&hist[(bits[i] >> shift) & 255], 1u);
    __syncthreads();
    // inclusive suffix sum over 256 buckets
    for (int off = 1; off < 256; off <<= 1) {
      const unsigned v = hist[tid];
      const unsigned w = (tid + off < 256) ? hist[tid + off] : 0u;
      __syncthreads();
      hist[tid] = v + w;
      __syncthreads();
    }
    const unsigned s     = hist[tid];
    const unsigned snext = (tid == 255) ? 0u : hist[tid + 1];
    if (s >= remaining && snext < remaining) {
      selB = (unsigned)tid;
      selR = remaining - snext;
    }
    __syncthreads();
    prefix |= (selB << shift);
    remaining = selR;
    __syncthreads();
  }

  // row-sum over kept entries (bits >= threshold)
  float s = 0.f;
#pragma unroll
  for (int i = 0; i < 8; ++i)
    if (bits[i] >= prefix) s += x[i];
  fsum[tid] = s;
  __syncthreads();
  for (int off = 128; off >= 1; off >>= 1) {
    if (tid < off) fsum[tid] += fsum[tid + off];
    __syncthreads();
  }
  if (tid == 0) {
    thr[rowIdx] = prefix;
    const float rs = fsum[0];
    dinv[rowIdx] = (rs > 0.f) ? rsqrtf(rs) : 0.f;
  }
}

// ---------------------------------------------------------------------------
// Kernel 4: in-place finalize: out = dinv_r * (x if kept else 0) * dinv_c.
// Streaming float4 pass over Adj -> NT load + NT store; dinv stays temporal
// (reused by every row of the batch).
// ---------------------------------------------------------------------------
__global__ __launch_bounds__(256) void sa_finalize_kernel(
    float* __restrict__ Adj, const float* __restrict__ mask,
    const unsigned* __restrict__ minkeys, const unsigned* __restrict__ thr,
    const float* __restrict__ dinv) {
  const int tid    = threadIdx.x;
  const int rowIdx = blockIdx.x;
  const int b      = rowIdx >> 11;
  const float minv = funkey(minkeys[b]);
  const float mr   = mask[rowIdx];
  const unsigned t = thr[rowIdx];
  const float dr   = dinv[rowIdx];

  v4f* __restrict__ row = (v4f*)(Adj + (size_t)rowIdx * N_);
  const v4f* __restrict__ dcol = (const v4f*)(dinv + (size_t)b * N_);

#pragma unroll
  for (int i = 0; i < 2; ++i) {
    const int c4 = i * 256 + tid;
    const v4f v = __builtin_nontemporal_load(&row[c4]);
    const v4f d = dcol[c4];
    v4f o;
    {
      const float xx = (v.x - minv) * mr;
      o.x = (__float_as_uint(xx) >= t) ? xx * dr * d.x : 0.f;
      const float xy = (v.y - minv) * mr;
      o.y = (__float_as_uint(xy) >= t) ? xy * dr * d.y : 0.f;
      const float xz = (v.z - minv) * mr;
      o.z = (__float_as_uint(xz) >= t) ? xz * dr * d.z : 0.f;
      const float xw = (v.w - minv) * mr;
      o.w = (__float_as_uint(xw) >= t) ? xw * dr * d.w : 0.f;
    }
    __builtin_nontemporal_store(o, &row[c4]);
  }
}

// ---------------------------------------------------------------------------
extern "C" void kernel_launch(void* const* d_in, const int* in_sizes, int n_in,
                              void* d_out, int out_size, void* d_ws,
                              size_t ws_size, hipStream_t stream) {
  (void)in_sizes; (void)n_in; (void)out_size; (void)ws_size;

  const float* ge   = (const float*)d_in[0];  // [16,2048,512]
  const float* mask = (const float*)d_in[1];  // [16,2048]
  const float* hs   = (const float*)d_in[2];  // [16,512]
  const float* W1   = (const float*)d_in[3];  // [1024,512]
  const float* b1   = (const float*)d_in[4];  // [512]
  const float* W2   = (const float*)d_in[5];  // [1024,512]
  const float* b2   = (const float*)d_in[6];  // [512]
  float* out = (float*)d_out;                 // [16,2048,2048]

  char* ws = (char*)d_ws;
  const size_t eElems = (size_t)B_ * N_ * OUT_;       // 16M floats
  float* e1 = (float*)ws;
  float* e2 = (float*)(ws + eElems * sizeof(float));
  unsigned* minkeys = (unsigned*)(ws + 2 * eElems * sizeof(float));
  unsigned* thr = (unsigned*)((char*)minkeys + 64);
  float* dinv = (float*)((char*)thr + (size_t)B_ * N_ * sizeof(unsigned));

  sa_init_kernel<<<1, 32, 0, stream>>>(minkeys);
  sa_embed_kernel<<<16384, 128, 0, stream>>>(ge, mask, hs, W1, b1, W2, b2, e1,
                                             e2);
  sa_adj_kernel<<<8192, 128, 0, stream>>>(e1, e2, out, minkeys);
  sa_topk_kernel<<<B_ * N_, 256, 0, stream>>>(out, mask, minkeys, thr, dinv);
  sa_finalize_kernel<<<B_ * N_, 256, 0, stream>>>(out, mask, minkeys, thr,
                                                  dinv);
}